// TemporalMemory_78984448573532
// MI455X (gfx1250) — compile-verified
//
#include <hip/hip_runtime.h>

typedef float v2f __attribute__((ext_vector_type(2)));
typedef float v4f __attribute__((ext_vector_type(4)));
typedef float v8f __attribute__((ext_vector_type(8)));

#define NUM_NODES 1000000
#define MEM_DIM 128
#define BATCH 262144

// ---------------- winner (dedup) kernels ----------------
__global__ void init_winner_kernel(int* __restrict__ w, int n) {
    for (int i = blockIdx.x * blockDim.x + threadIdx.x; i < n;
         i += gridDim.x * blockDim.x)
        w[i] = -1;
}

__global__ void mark_winner_kernel(const int* __restrict__ ids, int* __restrict__ w, int n) {
    for (int i = blockIdx.x * blockDim.x + threadIdx.x; i < n;
         i += gridDim.x * blockDim.x)
        atomicMax(&w[ids[i]], i);
}

// ---------------- bulk copy (b128, non-temporal both ways) ----------------
__global__ void copy4_kernel(const v4f* __restrict__ src, v4f* __restrict__ dst, long n) {
    long stride = (long)gridDim.x * blockDim.x;
    for (long i = (long)blockIdx.x * blockDim.x + threadIdx.x; i < n; i += stride) {
        v4f v = __builtin_nontemporal_load(&src[i]);
        __builtin_nontemporal_store(v, &dst[i]);
    }
}

// ---------------- fused GRU + scatter, f32 WMMA ----------------
__global__ __launch_bounds__(256)
void gru_wmma_kernel(const float* __restrict__ memory,
                     const float* __restrict__ messages,
                     const float* __restrict__ timestamps,
                     const float* __restrict__ W_ih,
                     const float* __restrict__ W_hh,
                     const float* __restrict__ b_ih,
                     const float* __restrict__ b_hh,
                     const int*   __restrict__ node_ids,
                     const int*   __restrict__ winner,
                     float* __restrict__ out_mem,
                     float* __restrict__ out_last)
{
    const int lane   = threadIdx.x & 31;
    const int wave   = threadIdx.x >> 5;
    const int m0     = (blockIdx.x * 8 + wave) * 16;   // first message row of this wave's tile

    const int rlocal = lane & 15;     // row (M) this lane serves in A fragments / N in B & C
    const int half   = lane >> 4;     // 0: K {0,1}; 1: K {2,3} within each K-block of 4
    const int koff   = half * 2;

    const int   mrow = m0 + rlocal;
    const int   node = node_ids[mrow];
    const int   win  = (winner[node] == mrow) ? 1 : 0;

    // scatter timestamps for winning rows (one lane per row); write-once -> NT
    if (lane < 16 && win)
        __builtin_nontemporal_store(timestamps[mrow], &out_last[node]);

    const float* xbase = messages + (long)mrow * MEM_DIM;
    const float* hbase = memory   + (long)node * MEM_DIM;

    #pragma unroll 1
    for (int t = 0; t < 8; ++t) {
        const int n0   = t * 16;
        const int wcol = n0 + rlocal;   // weight row (output unit) this lane supplies for B frags
        const float* wih_r = W_ih + (long)(      wcol) * MEM_DIM;
        const float* wih_z = W_ih + (long)(128 + wcol) * MEM_DIM;
        const float* wih_n = W_ih + (long)(256 + wcol) * MEM_DIM;
        const float* whh_r = W_hh + (long)(      wcol) * MEM_DIM;
        const float* whh_z = W_hh + (long)(128 + wcol) * MEM_DIM;
        const float* whh_n = W_hh + (long)(256 + wcol) * MEM_DIM;

        v8f air = {}, aiz = {}, ain = {}, ahr = {}, ahz = {}, ahn = {};

        #pragma unroll 4
        for (int kb = 0; kb < 32; ++kb) {
            const int ko = kb * 4 + koff;
            v2f ax = *(const v2f*)(xbase + ko);
            v2f ah = *(const v2f*)(hbase + ko);
            v2f br = *(const v2f*)(wih_r + ko);
            v2f bz = *(const v2f*)(wih_z + ko);
            v2f bn = *(const v2f*)(wih_n + ko);
            v2f cr = *(const v2f*)(whh_r + ko);
            v2f cz = *(const v2f*)(whh_z + ko);
            v2f cn = *(const v2f*)(whh_n + ko);
            air = __builtin_amdgcn_wmma_f32_16x16x4_f32(false, ax, false, br, (short)0, air, false, false);
            aiz = __builtin_amdgcn_wmma_f32_16x16x4_f32(false, ax, false, bz, (short)0, aiz, false, false);
            ain = __builtin_amdgcn_wmma_f32_16x16x4_f32(false, ax, false, bn, (short)0, ain, false, false);
            ahr = __builtin_amdgcn_wmma_f32_16x16x4_f32(false, ah, false, cr, (short)0, ahr, false, false);
            ahz = __builtin_amdgcn_wmma_f32_16x16x4_f32(false, ah, false, cz, (short)0, ahz, false, false);
            ahn = __builtin_amdgcn_wmma_f32_16x16x4_f32(false, ah, false, cn, (short)0, ahn, false, false);
        }

        // epilogue: C/D layout is M = e + 8*half, N = lane&15
        const int   c   = n0 + rlocal;
        const float bir = b_ih[c], biz = b_ih[128 + c], bin = b_ih[256 + c];
        const float bhr = b_hh[c], bhz = b_hh[128 + c], bhn = b_hh[256 + c];

        #pragma unroll
        for (int e = 0; e < 8; ++e) {
            const int   M      = e + half * 8;
            const int   node_e = __shfl(node, M, 32);
            const int   win_e  = __shfl(win,  M, 32);
            const float h_old  = memory[(long)node_e * MEM_DIM + c];

            const float xr = air[e] + bir + ahr[e] + bhr;
            const float xz = aiz[e] + biz + ahz[e] + bhz;
            const float r  = 1.0f / (1.0f + expf(-xr));
            const float z  = 1.0f / (1.0f + expf(-xz));
            const float nn = tanhf(ain[e] + bin + r * (ahn[e] + bhn));
            const float hn = (1.0f - z) * nn + z * h_old;

            if (win_e)
                __builtin_nontemporal_store(hn, &out_mem[(long)node_e * MEM_DIM + c]);
        }
    }
}

extern "C" void kernel_launch(void* const* d_in, const int* in_sizes, int n_in,
                              void* d_out, int out_size, void* d_ws, size_t ws_size,
                              hipStream_t stream) {
    const float* memory      = (const float*)d_in[0];
    const float* last_update = (const float*)d_in[1];
    const float* messages    = (const float*)d_in[2];
    const float* timestamps  = (const float*)d_in[3];
    const float* W_ih        = (const float*)d_in[4];
    const float* W_hh        = (const float*)d_in[5];
    const float* b_ih        = (const float*)d_in[6];
    const float* b_hh        = (const float*)d_in[7];
    const int*   node_ids    = (const int*)d_in[8];

    float* out_mem  = (float*)d_out;
    float* out_last = out_mem + (long)NUM_NODES * MEM_DIM;
    int*   winner   = (int*)d_ws;   // NUM_NODES ints (4 MB)

    // 1) dedup: last message index wins per node (deterministic)
    init_winner_kernel<<<2048, 256, 0, stream>>>(winner, NUM_NODES);
    mark_winner_kernel<<<1024, 256, 0, stream>>>(node_ids, winner, BATCH);

    // 2) bulk copy memory -> out, last_update -> out tail (b128, non-temporal)
    copy4_kernel<<<8192, 256, 0, stream>>>((const v4f*)memory, (v4f*)out_mem,
                                           (long)NUM_NODES * MEM_DIM / 4);
    copy4_kernel<<<512, 256, 0, stream>>>((const v4f*)last_update, (v4f*)out_last,
                                          NUM_NODES / 4);

    // 3) fused GRU (f32 WMMA) + scatter of winning rows
    gru_wmma_kernel<<<BATCH / 128, 256, 0, stream>>>(memory, messages, timestamps,
                                                     W_ih, W_hh, b_ih, b_hh,
                                                     node_ids, winner,
                                                     out_mem, out_last);
}